// DICE_50560355009142
// MI455X (gfx1250) — compile-verified
//
#include <hip/hip_runtime.h>
#include <hip/hip_bf16.h>

typedef __attribute__((ext_vector_type(2))) float v2f;
typedef __attribute__((ext_vector_type(8))) float v8f;

// ---------------------------------------------------------------- utilities
__device__ __forceinline__ float logsig_f(float x) {
  // log(sigmoid(x)) stable form
  float l = __logf(1.0f + __expf(-fabsf(x)));
  return (x >= 0.0f) ? -l : (x - l);
}

__device__ __forceinline__ void claim_slot(int* map, int r, int* cnt) {
  if (atomicCAS(&map[r], -1, -2) == -1) {
    int slot = atomicAdd(cnt, 1);
    atomicExch(&map[r], slot);
  }
}

__device__ __forceinline__ void mark_mask(int* m, int r, int* c) {
  if (atomicExch(&m[r], 1) == 0) atomicAdd(c, 1);
}

// ---------------------------------------------------------------- kernels
// Mark needed rows (compact slot allocation) + L1 membership masks.
__global__ void dice_mark_kernel(const int* __restrict__ user,
                                 const int* __restrict__ sp, const int* __restrict__ sn,
                                 const int* __restrict__ tp, const int* __restrict__ tn,
                                 int* mapS, int* mapT, int* cntS, int* cntT,
                                 int* mU, int* mS, int* mT,
                                 int* cU, int* cS, int* cT, int Bn, int Un) {
  int b = blockIdx.x * blockDim.x + threadIdx.x;
  if (b >= Bn) return;
  int u = user[b], p = sp[b], n = sn[b], q = tp[b], r = tn[b];
  claim_slot(mapS, u, cntS);
  claim_slot(mapS, Un + p, cntS);
  claim_slot(mapS, Un + n, cntS);
  claim_slot(mapT, u, cntT);
  claim_slot(mapT, Un + q, cntT);
  claim_slot(mapT, Un + r, cntT);
  mark_mask(mU, u, cU);
  mark_mask(mS, p, cS);
  mark_mask(mS, n, cS);
  mark_mask(mT, q, cT);
  mark_mask(mT, r, cT);
}

// Filtered COO aggregation for BOTH factors (int+pop) in one pass.
// Wave reads 32 edges coalesced, ballots survivors (~15%), broadcasts via
// shfl, then lanes scatter float2 slices with L2-resident f32 atomics.
__global__ void dice_scatter_kernel(const int* __restrict__ rows,
                                    const int* __restrict__ cols,
                                    const float* __restrict__ vals,
                                    const float* __restrict__ dri,
                                    const float* __restrict__ drp,
                                    const float* __restrict__ eUi, const float* __restrict__ eIi,
                                    const float* __restrict__ eUp, const float* __restrict__ eIp,
                                    const int* __restrict__ map,
                                    float* __restrict__ aggI, float* __restrict__ aggP,
                                    int nnz, int nU) {
  int tid = blockIdx.x * blockDim.x + threadIdx.x;
  int lane = tid & 31;
  int wave = tid >> 5;
  int nw = (gridDim.x * blockDim.x) >> 5;
  int o = lane * 2;
  for (int k0 = wave * 32; k0 < nnz; k0 += nw * 32) {
    __builtin_prefetch(rows + k0 + nw * 32, 0, 1);  // global_prefetch_b8
    int k = k0 + lane;
    int slot = -1, c = 0;
    float wi = 0.f, wp = 0.f;
    if (k < nnz) {
      int r = rows[k];          // coalesced 4B read for every edge
      slot = map[r];            // <0 => row never read downstream: skip edge
      if (slot >= 0) {
        c = cols[k];
        float v = vals[k];
        wi = v * dri[k];
        wp = v * drp[k];
      }
    }
    unsigned long long act = __ballot(slot >= 0);
    while (act) {
      int j = __ffsll((long long)act) - 1;
      act &= act - 1;
      int sj = __shfl(slot, j, 32);
      int cj = __shfl(c, j, 32);
      float wij = __shfl(wi, j, 32);
      float wpj = __shfl(wp, j, 32);
      const float* si = (cj < nU) ? (eUi + (size_t)cj * 64 + o)
                                  : (eIi + (size_t)(cj - nU) * 64 + o);
      const float* spp = (cj < nU) ? (eUp + (size_t)cj * 64 + o)
                                   : (eIp + (size_t)(cj - nU) * 64 + o);
      v2f ei = *(const v2f*)si;
      v2f ep = *(const v2f*)spp;
      float* di = aggI + (size_t)sj * 64 + o;
      float* dp = aggP + (size_t)sj * 64 + o;
      atomicAdd(di + 0, wij * ei.x);
      atomicAdd(di + 1, wij * ei.y);
      atomicAdd(dp + 0, wpj * ep.x);
      atomicAdd(dp + 1, wpj * ep.y);
    }
  }
}

// One domain (source or target): 16 batch elements per wave, 4 WMMA dot
// chains (pos/neg x int/pop). A = 16x4 product tile, B = e1 column so
// D[:,0] accumulates the 64-element dots; column 0 of C lives in lanes 0
// (M=0..7 across the 8 acc VGPRs) and 16 (M=8..15). Per-element BPR terms
// are separable across domains, so each launch adds into the same scalar.
__global__ void dice_dot_loss_dom(
    const int* __restrict__ user,
    const int* __restrict__ pos, const int* __restrict__ neg,
    const unsigned char* __restrict__ mask,
    const float* __restrict__ eUi, const float* __restrict__ eIi,
    const float* __restrict__ eUp, const float* __restrict__ eIp,
    const int* __restrict__ map,
    const float* __restrict__ aggI, const float* __restrict__ aggP,
    float* __restrict__ acc, int Bn, int Un, int nwaves) {
  int tid = blockIdx.x * blockDim.x + threadIdx.x;
  int g = tid >> 5;
  int lane = tid & 31;
  if (g >= nwaves) return;  // wave-uniform: EXEC stays all-ones for WMMA
  int m = lane & 15;
  int hh = lane >> 4;
  int koff = hh * 2;        // A layout: VGPR0<->K=k+2*(lane/16), VGPR1<->K+1
  int b0 = g * 16;
  int bm = b0 + m;
  if (bm >= Bn) bm = Bn - 1;  // clamp tail: duplicate work, masked at reduce

  int u = user[bm], p = pos[bm], n = neg[bm];
  const float* pEUi = eUi + (size_t)u * 64;
  const float* pEUp = eUp + (size_t)u * 64;
  const float* pAUi = aggI + (size_t)map[u] * 64;
  const float* pAUp = aggP + (size_t)map[u] * 64;
  const float* pEPi = eIi + (size_t)p * 64;
  const float* pENi = eIi + (size_t)n * 64;
  const float* pEPp = eIp + (size_t)p * 64;
  const float* pENp = eIp + (size_t)n * 64;
  const float* pAPi = aggI + (size_t)map[Un + p] * 64;
  const float* pANi = aggI + (size_t)map[Un + n] * 64;
  const float* pAPp = aggP + (size_t)map[Un + p] * 64;
  const float* pANp = aggP + (size_t)map[Un + n] * 64;

  float bsel = (m == 0) ? 1.0f : 0.0f;  // B e1 column: B[k][0]=1
  v2f bvec;
  bvec.x = bsel;
  bvec.y = bsel;
  v8f dPi = {0.f, 0.f, 0.f, 0.f, 0.f, 0.f, 0.f, 0.f};
  v8f dNi = dPi, dPp = dPi, dNp = dPi;

#pragma unroll
  for (int k = 0; k < 64; k += 4) {
    int o = k + koff;
    // user slices shared by pos & neg chains of each factor
    v2f ui = 0.5f * (*(const v2f*)(pEUi + o) + *(const v2f*)(pAUi + o));
    v2f up = 0.5f * (*(const v2f*)(pEUp + o) + *(const v2f*)(pAUp + o));
    v2f xpi = 0.5f * (*(const v2f*)(pEPi + o) + *(const v2f*)(pAPi + o));
    v2f xni = 0.5f * (*(const v2f*)(pENi + o) + *(const v2f*)(pANi + o));
    v2f xpp = 0.5f * (*(const v2f*)(pEPp + o) + *(const v2f*)(pAPp + o));
    v2f xnp = 0.5f * (*(const v2f*)(pENp + o) + *(const v2f*)(pANp + o));
    dPi = __builtin_amdgcn_wmma_f32_16x16x4_f32(false, ui * xpi, false, bvec,
                                                (short)0, dPi, false, false);
    dNi = __builtin_amdgcn_wmma_f32_16x16x4_f32(false, ui * xni, false, bvec,
                                                (short)0, dNi, false, false);
    dPp = __builtin_amdgcn_wmma_f32_16x16x4_f32(false, up * xpp, false, bvec,
                                                (short)0, dPp, false, false);
    dNp = __builtin_amdgcn_wmma_f32_16x16x4_f32(false, up * xnp, false, bvec,
                                                (short)0, dNp, false, false);
  }

  if (m == 0) {  // lanes 0 and 16 hold C column 0: M=0..7 / M=8..15
    float ls = 0.f;
#define DICE_ACCJ(j)                                                       \
  {                                                                        \
    int bb = b0 + hh * 8 + (j);                                            \
    if (bb < Bn) {                                                         \
      float pi = dPi[j], ni = dNi[j], pp = dPp[j], np2 = dNp[j];           \
      bool mk = mask[bb] != 0;                                             \
      float c = logsig_f((pi + pp) - (ni + np2));                          \
      c += 0.1f * (mk ? logsig_f(pi - ni) : 0.f);                          \
      c += 0.1f * (mk ? logsig_f(np2 - pp) : logsig_f(pp - np2));          \
      ls += c;                                                             \
    }                                                                      \
  }
    DICE_ACCJ(0) DICE_ACCJ(1) DICE_ACCJ(2) DICE_ACCJ(3)
    DICE_ACCJ(4) DICE_ACCJ(5) DICE_ACCJ(6) DICE_ACCJ(7)
#undef DICE_ACCJ
    atomicAdd(acc, -ls);
  }
}

// Masked L1 sum over flagged rows: one wave per row, float2 slices.
__global__ void dice_l1_kernel(const float* __restrict__ a, const float* __restrict__ b,
                               const int* __restrict__ mask, float* __restrict__ acc,
                               int nrows) {
  int tid = blockIdx.x * blockDim.x + threadIdx.x;
  int lane = tid & 31;
  int wave = tid >> 5;
  int nw = (gridDim.x * blockDim.x) >> 5;
  int o = lane * 2;
  float s = 0.f;
  for (int r = wave; r < nrows; r += nw) {
    if (!mask[r]) continue;
    v2f x = *(const v2f*)(a + (size_t)r * 64 + o);
    v2f y = *(const v2f*)(b + (size_t)r * 64 + o);
    s += fabsf(x.x - y.x) + fabsf(x.y - y.y);
  }
  for (int off = 16; off > 0; off >>= 1) s += __shfl_xor(s, off, 32);
  if (lane == 0 && s != 0.f) atomicAdd(acc, s);
}

__global__ void dice_final_kernel(const int* __restrict__ ci, const float* __restrict__ cf,
                                  float* __restrict__ out, int Bn) {
  int cU = ci[2] > 1 ? ci[2] : 1;
  int cS = ci[3] > 1 ? ci[3] : 1;
  int cT = ci[4] > 1 ? ci[4] : 1;
  float dis = cf[1] / (cU * 64.0f) + cf[2] / (cS * 64.0f) +
              cf[3] / (cU * 64.0f) + cf[4] / (cT * 64.0f);
  out[0] = cf[0] / (float)Bn - 0.01f * dis;
}

// ---------------------------------------------------------------- launch
extern "C" void kernel_launch(void* const* d_in, const int* in_sizes, int n_in,
                              void* d_out, int out_size, void* d_ws, size_t ws_size,
                              hipStream_t stream) {
  const int* user = (const int*)d_in[0];
  const int* spos = (const int*)d_in[1];
  const int* sneg = (const int*)d_in[2];
  const int* tpos = (const int*)d_in[3];
  const int* tneg = (const int*)d_in[4];
  const unsigned char* maskS = (const unsigned char*)d_in[5];
  const unsigned char* maskT = (const unsigned char*)d_in[6];
  const float* suI = (const float*)d_in[7];
  const float* siI = (const float*)d_in[8];
  const float* tuI = (const float*)d_in[9];
  const float* tiI = (const float*)d_in[10];
  const float* suP = (const float*)d_in[11];
  const float* siP = (const float*)d_in[12];
  const float* tuP = (const float*)d_in[13];
  const float* tiP = (const float*)d_in[14];
  const int* s_rows = (const int*)d_in[15];
  const int* s_cols = (const int*)d_in[16];
  const float* s_vals = (const float*)d_in[17];
  const int* t_rows = (const int*)d_in[18];
  const int* t_cols = (const int*)d_in[19];
  const float* t_vals = (const float*)d_in[20];
  const float* drop_s_int = (const float*)d_in[21];
  const float* drop_t_int = (const float*)d_in[22];
  const float* drop_s_pop = (const float*)d_in[23];
  const float* drop_t_pop = (const float*)d_in[24];

  const int Bn = in_sizes[0];
  const int Un = in_sizes[7] / 64;
  const int SIn = in_sizes[8] / 64;
  const int TIn = in_sizes[10] / 64;
  const int NNZS = in_sizes[15];
  const int NNZT = in_sizes[18];
  const int Ns = Un + SIn, Nt = Un + TIn;
  const int CAP = 3 * Bn;  // max distinct needed rows per domain

  char* w = (char*)d_ws;
  size_t off = 0;
  auto take = [&](size_t bytes) -> void* {
    void* p = (void*)(w + off);
    off += (bytes + 255) & ~(size_t)255;
    return p;
  };
  float* aggSI = (float*)take((size_t)CAP * 64 * 4);
  float* aggSP = (float*)take((size_t)CAP * 64 * 4);
  float* aggTI = (float*)take((size_t)CAP * 64 * 4);
  float* aggTP = (float*)take((size_t)CAP * 64 * 4);
  int* mapS = (int*)take((size_t)Ns * 4);
  int* mapT = (int*)take((size_t)Nt * 4);
  int* mU = (int*)take((size_t)Un * 4);
  int* mS = (int*)take((size_t)SIn * 4);
  int* mT = (int*)take((size_t)TIn * 4);
  int* ctrl = (int*)take(256);
  int* ci = ctrl;                  // [0]=cntS [1]=cntT [2]=cU [3]=cS [4]=cT
  float* cf = (float*)(ctrl + 8);  // [0]=bpr sum, [1..4]=L1 sums

  // deterministic init (graph-capture safe)
  hipMemsetAsync(aggSI, 0, (size_t)CAP * 256 * 4, stream);  // 4 contiguous agg bufs
  hipMemsetAsync(mapS, 0xFF, (size_t)Ns * 4, stream);
  hipMemsetAsync(mapT, 0xFF, (size_t)Nt * 4, stream);
  hipMemsetAsync(mU, 0, (size_t)Un * 4, stream);
  hipMemsetAsync(mS, 0, (size_t)SIn * 4, stream);
  hipMemsetAsync(mT, 0, (size_t)TIn * 4, stream);
  hipMemsetAsync(ctrl, 0, 256, stream);

  dice_mark_kernel<<<(Bn + 255) / 256, 256, 0, stream>>>(
      user, spos, sneg, tpos, tneg, mapS, mapT, ci + 0, ci + 1,
      mU, mS, mT, ci + 2, ci + 3, ci + 4, Bn, Un);

  dice_scatter_kernel<<<1024, 256, 0, stream>>>(
      s_rows, s_cols, s_vals, drop_s_int, drop_s_pop,
      suI, siI, suP, siP, mapS, aggSI, aggSP, NNZS, Un);
  dice_scatter_kernel<<<1024, 256, 0, stream>>>(
      t_rows, t_cols, t_vals, drop_t_int, drop_t_pop,
      tuI, tiI, tuP, tiP, mapT, aggTI, aggTP, NNZT, Un);

  int nwaves = (Bn + 15) / 16;
  int nblk = (nwaves * 32 + 255) / 256;
  dice_dot_loss_dom<<<nblk, 256, 0, stream>>>(
      user, spos, sneg, maskS, suI, siI, suP, siP,
      mapS, aggSI, aggSP, cf + 0, Bn, Un, nwaves);
  dice_dot_loss_dom<<<nblk, 256, 0, stream>>>(
      user, tpos, tneg, maskT, tuI, tiI, tuP, tiP,
      mapT, aggTI, aggTP, cf + 0, Bn, Un, nwaves);

  dice_l1_kernel<<<256, 256, 0, stream>>>(suI, suP, mU, cf + 1, Un);
  dice_l1_kernel<<<256, 256, 0, stream>>>(siI, siP, mS, cf + 2, SIn);
  dice_l1_kernel<<<256, 256, 0, stream>>>(tuI, tuP, mU, cf + 3, Un);
  dice_l1_kernel<<<256, 256, 0, stream>>>(tiI, tiP, mT, cf + 4, TIn);

  dice_final_kernel<<<1, 1, 0, stream>>>(ci, cf, (float*)d_out, Bn);
}